// DANetwork_635655159916
// MI455X (gfx1250) — compile-verified
//
#include <hip/hip_runtime.h>

// ---------------------------------------------------------------------------
// Problem constants (from reference): x is (B=16, C=48, H=128, W=128) f32.
// Output is (16, 192, 128, 128) f32 = [s2|s3|s4|s5] along channels.
// ---------------------------------------------------------------------------
#define Bn 16
#define Cn 48
#define Hn 128
#define Wn 128
#define HW (Hn * Wn)

typedef __attribute__((ext_vector_type(2))) float v2f;
typedef __attribute__((ext_vector_type(8))) float v8f;

// reflect index for pad=1, 'reflect' (edge not repeated): -1 -> 1, 128 -> 126
__device__ __forceinline__ int refl(int i) {
  i = (i < 0) ? -i : i;
  return (i > Wn - 1) ? (2 * Wn - 2 - i) : i;
}

// CDNA5 async global->LDS copy (ASYNCcnt path). LDS dest offset in a VGPR,
// 64-bit global address in a VGPR pair, saddr = off.
__device__ __forceinline__ void async_ld_f32(unsigned lds_off, const float* g) {
  asm volatile("global_load_async_to_lds_b32 %0, %1, off"
               :: "v"(lds_off), "v"(g) : "memory");
}
__device__ __forceinline__ void wait_async0() {
  asm volatile("s_wait_asynccnt 0x0" ::: "memory");
}

// ---------------------------------------------------------------------------
// Pass 1: per (channel, batch) partial sums for mean/var.
// ws layout (floats): [0..1535] partials (c*16+b)*2 = {sum, sumsq}
//                     [1536..1583] mean, [1584..1631] invstd, [1632..1729] w[14][7]
// ---------------------------------------------------------------------------
__global__ __launch_bounds__(256) void stats_partial_kernel(
    const float* __restrict__ x, float* __restrict__ part) {
  const int c = blockIdx.x, b = blockIdx.y, t = threadIdx.x;
  const float* p = x + (((b * Cn) + c) << 14);
  float s = 0.f, q = 0.f;
  for (int i = t; i < HW; i += 256) {
    float v = p[i];
    s += v;
    q += v * v;
  }
  __shared__ float sA[256], sB[256];
  sA[t] = s; sB[t] = q;
  __syncthreads();
  for (int o = 128; o > 0; o >>= 1) {
    if (t < o) { sA[t] += sA[t + o]; sB[t] += sB[t + o]; }
    __syncthreads();
  }
  if (t == 0) {
    part[(c * Bn + b) * 2 + 0] = sA[0];
    part[(c * Bn + b) * 2 + 1] = sB[0];
  }
}

// ---------------------------------------------------------------------------
// Pass 2: finalize mean/invstd per channel; softmax(alphas_reduce) rows.
// stats[0..47]=mean, [48..95]=invstd, [96..193]=w (14x7)
// ---------------------------------------------------------------------------
__global__ void finalize_kernel(const float* __restrict__ part,
                                const float* __restrict__ alphas,
                                float* __restrict__ stats) {
  const int t = threadIdx.x;
  if (t < Cn) {
    float S = 0.f, Q = 0.f;
    for (int b = 0; b < Bn; ++b) {
      S += part[(t * Bn + b) * 2 + 0];
      Q += part[(t * Bn + b) * 2 + 1];
    }
    const float N = (float)(Bn * HW);
    float m = S / N;
    float var = Q / N - m * m;
    stats[t] = m;
    stats[Cn + t] = rsqrtf(var + 1e-5f);
  }
  if (t < 14) {
    float a[7], mx = -1e30f;
    for (int j = 0; j < 7; ++j) { a[j] = alphas[t * 7 + j]; mx = fmaxf(mx, a[j]); }
    float s = 0.f;
    for (int j = 0; j < 7; ++j) { a[j] = expf(a[j] - mx); s += a[j]; }
    for (int j = 0; j < 7; ++j) stats[96 + t * 7 + j] = a[j] / s;
  }
}

// ---------------------------------------------------------------------------
// Pass 3..6: compute state s_step (step = 2..5) into its channel block of out.
// Block = 256 threads = 8 waves; block covers rows [i0, i0+16) x all 128 cols
// of one (b, c). Wave w owns the 16x16 tile at columns [16w, 16w+16).
// LDS holds a reflect-padded 18x130 stripe of the current source (raw x for
// source 0 — affine folded into coefficients; prior states read from out).
// Vertical blur leg = 5x v_wmma_f32_16x16x4_f32; D layout defines the
// thread->pixel map for all scalar terms and the store.
// ---------------------------------------------------------------------------
__global__ __launch_bounds__(256) void state_kernel(
    const float* __restrict__ x, const float* __restrict__ gamma,
    const float* __restrict__ beta, const float* __restrict__ stats,
    float* __restrict__ out, int step) {
  __shared__ float stripe[18 * 132];  // 18 rows x 130 cols (stride 132)

  const int t = threadIdx.x;
  const int lane = t & 31, wid = t >> 5;
  const int half = lane >> 4, nl = lane & 15;
  const int n0 = wid << 4;                    // tile column base
  const int ri = blockIdx.x, c = blockIdx.y, b = blockIdx.z;
  const int i0 = ri << 4;

  const int offs = (step == 2) ? 0 : (step == 3) ? 2 : (step == 4) ? 5 : 9;
  const int nsrc = step - 1;
  const float* wrow = stats + 96;
  const float meanc = stats[c], invc = stats[Cn + c];
  const float gam = gamma[c], bet = beta[c];

  v8f acc = {0.f, 0.f, 0.f, 0.f, 0.f, 0.f, 0.f, 0.f};

  for (int m = 0; m < nsrc; ++m) {
    __syncthreads();  // previous source's LDS reads done before overwrite

    const float* src;
    int base;
    float aff_a, aff_b;
    if (m == 0) {  // source is s = a*x + b (batchnorm folded)
      src = x;
      base = ((b * Cn) + c) << 14;
      aff_a = invc * gam;
      aff_b = bet - meanc * aff_a;
    } else {       // source is s_{m+1}, already in out channel block (m-1)
      src = out;
      base = ((b * 192) + 48 * (m - 1) + c) << 14;
      aff_a = 1.f;
      aff_b = 0.f;
    }
    float u[7];
    if (m == 0) {
      for (int j = 0; j < 7; ++j)
        u[j] = wrow[offs * 7 + j] + wrow[(offs + 1) * 7 + j];  // s0==s1 merge
    } else {
      for (int j = 0; j < 7; ++j) u[j] = wrow[(offs + 1 + m) * 7 + j];
    }
    const float c_id = aff_a * (u[0] - u[2]);
    const float c_bl = aff_a * u[1] * (1.f / 9.f);
    const float c_fw = aff_a * u[3];
    const float c_fh = aff_a * u[4];
    const float c_rl = aff_a * u[5];
    const float c_rt = aff_a * u[6];
    const float c_k =
        aff_b * (u[0] + u[1] + u[3] + u[4] + u[5] + u[6] - u[2]) + u[2];

    // ---- async global->LDS: 18x130 reflect-padded stripe (2340 elems) ----
    for (int e = t; e < 2560; e += 256) {  // uniform trip count, no divergence
      int e2 = (e < 2340) ? e : 2339;      // clamped dup writes are benign
      int r = e2 / 130, cc = e2 - r * 130;
      int grow = refl(i0 - 1 + r);
      int gcol = refl(cc - 1);
      unsigned lo = (unsigned)(unsigned long long)(&stripe[r * 132 + cc]);
      async_ld_f32(lo, src + base + (grow << 7) + gcol);
    }
    wait_async0();
    __syncthreads();

    // ---- vertical blur via WMMA: acc += A(16x18 band of 1s) x (c_bl*Hsum) --
    // A layout (32b 16x4): lanes0-15 M=lane {K=kb+0,kb+1}; lanes16-31 M=lane-16
    // {K=kb+2,kb+3}. B assumed symmetric: row K striped across lanes (N=lane%16).
    for (int kb = 0; kb < 20; kb += 4) {
      const int k0 = kb + (half ? 2 : 0);
      const int k1 = k0 + 1;
      const int M = nl;
      v2f av, bv;
      av.x = (k0 >= M && k0 <= M + 2 && k0 < 18) ? 1.f : 0.f;
      av.y = (k1 >= M && k1 <= M + 2 && k1 < 18) ? 1.f : 0.f;
      const int cb = n0 + nl;  // padded cols cb..cb+2 == global cols j-1..j+1
      bv.x = (k0 < 18) ? (stripe[k0 * 132 + cb] + stripe[k0 * 132 + cb + 1] +
                          stripe[k0 * 132 + cb + 2]) * c_bl
                       : 0.f;
      bv.y = (k1 < 18) ? (stripe[k1 * 132 + cb] + stripe[k1 * 132 + cb + 1] +
                          stripe[k1 * 132 + cb + 2]) * c_bl
                       : 0.f;
      acc = __builtin_amdgcn_wmma_f32_16x16x4_f32(false, av, false, bv,
                                                  (short)0, acc, false, false);
    }

    // ---- scalar terms, mapped onto the WMMA D layout (M=r|r+8, N=nl) ----
    for (int r = 0; r < 8; ++r) {
      const int M = r + (half ? 8 : 0);
      const int gi = i0 + M;
      const int gj = n0 + nl;
      const float idv = stripe[(M + 1) * 132 + gj + 1];    // identity (LDS)
      const float fwv = stripe[(M + 1) * 132 + (128 - gj)];// flip-W (LDS)
      const float fhv = src[base + ((127 - gi) << 7) + gj];               // flip-H
      const float rlv = src[base + (((gi + 124) & 127) << 7) + ((gj + 124) & 127)]; // roll(4,4)
      const float rtv = src[base + (gj << 7) + (127 - gi)];               // rot90
      acc[r] += c_id * idv + c_fw * fwv + c_fh * fhv + c_rl * rlv +
                c_rt * rtv + c_k;
    }
  }

  // ---- store into this state's channel block of out ----
  const int cbase = ((b * 192) + 48 * (step - 2) + c) << 14;
  for (int r = 0; r < 8; ++r) {
    const int gi = i0 + r + (half ? 8 : 0);
    out[cbase + (gi << 7) + n0 + nl] = acc[r];
  }
}

// ---------------------------------------------------------------------------
extern "C" void kernel_launch(void* const* d_in, const int* in_sizes, int n_in,
                              void* d_out, int out_size, void* d_ws,
                              size_t ws_size, hipStream_t stream) {
  const float* x = (const float*)d_in[0];
  const float* gamma = (const float*)d_in[1];
  const float* beta = (const float*)d_in[2];
  // d_in[3] = alphas_normal (unused by reference path)
  const float* areduce = (const float*)d_in[4];
  float* out = (float*)d_out;
  float* ws = (float*)d_ws;
  float* part = ws;            // 1536 floats
  float* stats = ws + 1536;    // 48 mean + 48 invstd + 98 softmax weights

  stats_partial_kernel<<<dim3(Cn, Bn), 256, 0, stream>>>(x, part);
  finalize_kernel<<<1, 64, 0, stream>>>(part, areduce, stats);
  for (int step = 2; step <= 5; ++step) {
    state_kernel<<<dim3(Hn / 16, Cn, Bn), 256, 0, stream>>>(x, gamma, beta,
                                                            stats, out, step);
  }
}